// SubglacialDrainageSystem_69363721830942
// MI455X (gfx1250) — compile-verified
//
#include <hip/hip_runtime.h>
#include <stdint.h>
#include <math.h>

// ---------------- Fixed problem geometry (from reference) ----------------
#define NXc 2048
#define NYc 2048
#define Nn  (NXc * NYc)                 // 4,194,304 nodes
#define LHc (NYc * (NXc - 1))           // 4,192,256 horizontal links
#define LVc ((NYc - 1) * NXc)           // 4,192,256 vertical links

typedef float f32x4 __attribute__((ext_vector_type(4)));

static constexpr size_t L_ALL   = (size_t)LHc + (size_t)LVc;      // 8,384,512
static constexpr size_t OFF_FLOW  = 0;
static constexpr size_t OFF_INOUT = L_ALL;
static constexpr size_t OFF_SQ    = L_ALL + (size_t)Nn;
static constexpr size_t OFF_CQ    = 2 * L_ALL + (size_t)Nn;
static constexpr size_t OFF_OPEN  = 3 * L_ALL + (size_t)Nn;
static constexpr size_t OFF_SC    = 3 * L_ALL + 2 * (size_t)Nn;
static constexpr size_t OFF_CC    = 3 * L_ALL + 3 * (size_t)Nn;
static constexpr size_t OFF_DISS  = 4 * L_ALL + 3 * (size_t)Nn;
static constexpr size_t OFF_SENS  = 5 * L_ALL + 3 * (size_t)Nn;

// ---------------- Physics constants ----------------
static constexpr float RHOW_G  = 1000.0f * 9.81f;          // 9810
static constexpr float RHOI_G  = 917.0f  * 9.81f;          // 8995.77
static constexpr float INV_LL  = 1.0f / 100.0f;            // 1/LINK_LEN
static constexpr float INV_SEC = 1.0f / 31556926.0f;
static constexpr float K_Sf    = 0.01f;
static constexpr float K_Cf    = 0.1f;
static constexpr float ACLf    = 5e-25f;
static constexpr float CTWf    = 7.5e-8f * 4220.0f * 1000.0f;  // 0.3165

__device__ __forceinline__ float relu3(float x) {
  float r = x > 0.0f ? x : 0.0f;
  return r * r * r;
}

struct LinkOut { float flow, sq, cq, cc, diss, sens; };

// Shared per-link math; returns the 6 link outputs.
__device__ __forceinline__ LinkOut link_math(float pt, float ph,
                                             float sht, float shh,
                                             float bt, float bh,
                                             float it, float ih,
                                             float c) {
  LinkOut r;
  const float grad = (ph - pt) * INV_LL;
  r.flow = (ph > pt) ? -1.0f : 1.0f;
  const float hl = 0.5f * (sht + shh);
  // h^1.25 = h * sqrt(sqrt(h));  |grad|^-0.5 = rsqrt(|grad|)
  r.sq = -K_Sf * (hl * sqrtf(sqrtf(hl))) * rsqrtf(fabsf(grad)) * grad;
  r.cq = -K_Cf * (c * c * c) * grad;
  const float base_t = RHOW_G * bt;
  const float base_h = RHOW_G * bh;
  const float eff_t  = base_t + RHOI_G * it - pt;
  const float eff_h  = base_h + RHOI_G * ih - ph;
  const float Nl     = 0.5f * (eff_t + eff_h);
  r.cc   = ACLf * c * relu3(Nl);
  r.diss = fabsf(r.cq * grad) + 2.0f * fabsf(r.sq * grad);  // L_C = 2
  const float pgrad = ((ph - base_h) - (pt - base_t)) * INV_LL;
  const float tq = ((c > 0.0f) || (pgrad * r.sq > 0.0f)) ? (r.cq + 2.0f * r.sq)
                                                         : r.cq;
  r.sens = -CTWf * tq * pgrad;
  return r;
}

// gfx1250 async global->LDS copies (ASYNCcnt path). lds_off = byte offset of
// the LDS destination inside the workgroup allocation; goff = signed 32-bit
// byte offset from the uniform 64-bit SGPR base (GVS mode).
#define ASYNC_G2L_B32(lds_off, goff, base64)                                   \
  asm volatile("global_load_async_to_lds_b32 %0, %1, %2"                       \
               :: "v"(lds_off), "v"(goff), "s"(base64) : "memory")
#define ASYNC_G2L_B128(lds_off, goff, base64)                                  \
  asm volatile("global_load_async_to_lds_b128 %0, %1, %2"                      \
               :: "v"(lds_off), "v"(goff), "s"(base64) : "memory")

// ---------------- Horizontal links: LDS-staged row tiles ----------------
// grid = (8, 2048), block = 256. Block (bx,y) covers links x in
// [bx*256, min(bx*256+256, 2047)), needing nodes [bx*256, bx*256+nlinks].
// Staging: lane t copies 16B of array (t>>6) via one async b128; the array
// selector is uniform per wave32, so each asm sees a uniform SGPR base.
__global__ __launch_bounds__(256) void hlink_kernel(
    const float* __restrict__ pot,  const float* __restrict__ sheet,
    const float* __restrict__ cs,   const float* __restrict__ bed,
    const float* __restrict__ ice,  float* __restrict__ out) {
  // rows: 0=pot 1=sheet 2=bed 3=ice; stride 260 floats keeps rows 16B-aligned
  __shared__ __align__(16) float tiles[4][260];

  const int y  = blockIdx.y;
  const int x0 = blockIdx.x * 256;
  const int t  = threadIdx.x;
  const int nlinks  = min(256, (NXc - 1) - x0);   // 256 or 255 (last block)
  const int rowbase = y * NXc + x0;

  const uint64_t bpot   = (uint64_t)(uintptr_t)pot;
  const uint64_t bsheet = (uint64_t)(uintptr_t)sheet;
  const uint64_t bbed   = (uint64_t)(uintptr_t)bed;
  const uint64_t bice   = (uint64_t)(uintptr_t)ice;

  {
    const int a = t >> 6;            // wave-uniform array selector
    const int l = t & 63;            // 16B chunk within the 256-node tile
    const int goff = (rowbase + 4 * l) * 4;
    const unsigned ldsoff = (unsigned)(uintptr_t)&tiles[a][4 * l];
    switch (a) {
      case 0: ASYNC_G2L_B128(ldsoff, goff, bpot);   break;
      case 1: ASYNC_G2L_B128(ldsoff, goff, bsheet); break;
      case 2: ASYNC_G2L_B128(ldsoff, goff, bbed);   break;
      default: ASYNC_G2L_B128(ldsoff, goff, bice);  break;
    }
  }
  // one extra node (index 256) when the block has a full 256 links.
  if (nlinks == 256 && t == 0) {
    const int goff = (rowbase + 256) * 4;
    ASYNC_G2L_B32((unsigned)(uintptr_t)&tiles[0][256], goff, bpot);
    ASYNC_G2L_B32((unsigned)(uintptr_t)&tiles[1][256], goff, bsheet);
    ASYNC_G2L_B32((unsigned)(uintptr_t)&tiles[2][256], goff, bbed);
    ASYNC_G2L_B32((unsigned)(uintptr_t)&tiles[3][256], goff, bice);
  }
  asm volatile("s_wait_asynccnt 0x0" ::: "memory");
  __syncthreads();

  if (t < nlinks) {
    const size_t link = (size_t)y * (NXc - 1) + (size_t)(x0 + t);
    const float c = __builtin_nontemporal_load(cs + link);   // read-once stream
    const LinkOut r = link_math(tiles[0][t], tiles[0][t + 1],
                                tiles[1][t], tiles[1][t + 1],
                                tiles[2][t], tiles[2][t + 1],
                                tiles[3][t], tiles[3][t + 1], c);
    // write-once outputs: NT stores keep the shared inputs resident in L2
    __builtin_nontemporal_store(r.flow, out + OFF_FLOW + link);
    __builtin_nontemporal_store(r.sq,   out + OFF_SQ   + link);
    __builtin_nontemporal_store(r.cq,   out + OFF_CQ   + link);
    __builtin_nontemporal_store(r.cc,   out + OFF_CC   + link);
    __builtin_nontemporal_store(r.diss, out + OFF_DISS + link);
    __builtin_nontemporal_store(r.sens, out + OFF_SENS + link);
  }
}

// ---------------- Vertical links: b128 streaming, 4 links/thread ----------
// LV = 4,192,256 = 4094 blocks * 256 threads * 4 links. Fully coalesced,
// all 16B accesses aligned (every offset divisible by 4 elements).
__global__ __launch_bounds__(256) void vlink_kernel(
    const float* __restrict__ pot,  const float* __restrict__ sheet,
    const float* __restrict__ cs,   const float* __restrict__ bed,
    const float* __restrict__ ice,  float* __restrict__ out) {
  const int j0 = 4 * (blockIdx.x * 256 + threadIdx.x);   // tail node / v-link idx

  // cover latency of the +NXc strided streams (speculative prefetch is safe).
  __builtin_prefetch(pot + j0 + NXc + 8192, 0, 1);
  __builtin_prefetch(ice + j0 + NXc + 8192, 0, 1);

  const f32x4 pt = *(const f32x4*)(pot   + j0);
  const f32x4 ph = *(const f32x4*)(pot   + j0 + NXc);
  const f32x4 st = *(const f32x4*)(sheet + j0);
  const f32x4 sh = *(const f32x4*)(sheet + j0 + NXc);
  const f32x4 bt = *(const f32x4*)(bed   + j0);
  const f32x4 bh = *(const f32x4*)(bed   + j0 + NXc);
  const f32x4 it = *(const f32x4*)(ice   + j0);
  const f32x4 ih = *(const f32x4*)(ice   + j0 + NXc);
  const f32x4 cc = __builtin_nontemporal_load(
      (const f32x4*)(cs + (size_t)LHc + j0));            // read-once stream

  f32x4 vfl, vsq, vcq, vcc, vdi, vse;
#pragma unroll
  for (int k = 0; k < 4; ++k) {
    const LinkOut r = link_math(pt[k], ph[k], st[k], sh[k],
                                bt[k], bh[k], it[k], ih[k], cc[k]);
    vfl[k] = r.flow; vsq[k] = r.sq; vcq[k] = r.cq;
    vcc[k] = r.cc;   vdi[k] = r.diss; vse[k] = r.sens;
  }

  const size_t link0 = (size_t)LHc + (size_t)j0;
  __builtin_nontemporal_store(vfl, (f32x4*)(out + OFF_FLOW + link0));
  __builtin_nontemporal_store(vsq, (f32x4*)(out + OFF_SQ   + link0));
  __builtin_nontemporal_store(vcq, (f32x4*)(out + OFF_CQ   + link0));
  __builtin_nontemporal_store(vcc, (f32x4*)(out + OFF_CC   + link0));
  __builtin_nontemporal_store(vdi, (f32x4*)(out + OFF_DISS + link0));
  __builtin_nontemporal_store(vse, (f32x4*)(out + OFF_SENS + link0));
}

// ---------------- Nodes: inflow/outflow, opening, sheet closure -----------
// grid = (8, 2048), block = 256. Neighbor gathers are row-local (L2-hot after
// the link kernels thanks to NT output stores); the segment_sum is replaced
// by a direct gather of the <=4 incident links.
__global__ __launch_bounds__(256) void node_kernel(
    const float* __restrict__ pot,  const float* __restrict__ sheet,
    const float* __restrict__ bed,  const float* __restrict__ ice,
    const float* __restrict__ sv,   const int* __restrict__ status,
    float* __restrict__ out) {
  const int x = blockIdx.x * 256 + threadIdx.x;
  const int y = blockIdx.y;
  const int i = y * NXc + x;

  const float p = pot[i];
  float m = -INFINITY;
  if (x > 0)        m = fmaxf(m, pot[i - 1]);
  if (x < NXc - 1)  m = fmaxf(m, pot[i + 1]);
  if (y > 0)        m = fmaxf(m, pot[i - NXc]);
  if (y < NYc - 1)  m = fmaxf(m, pot[i + NXc]);

  const float b  = (__builtin_nontemporal_load(status + i) > 0) ? 1.0f : 0.0f;
  const float io = (p > m) ? b : -b;

  // mean |sliding_velocity|/SEC_PER_A over incident links (read-once stream)
  float num = 0.0f, cnt = 0.0f;
  if (x > 0)       { num += fabsf(__builtin_nontemporal_load(sv + y * (NXc - 1) + x - 1));   cnt += 1.0f; }
  if (x < NXc - 1) { num += fabsf(__builtin_nontemporal_load(sv + y * (NXc - 1) + x));       cnt += 1.0f; }
  if (y > 0)       { num += fabsf(__builtin_nontemporal_load(sv + LHc + (y - 1) * NXc + x)); cnt += 1.0f; }
  if (y < NYc - 1) { num += fabsf(__builtin_nontemporal_load(sv + LHc + y * NXc + x));       cnt += 1.0f; }
  const float snode = (num * INV_SEC) / fmaxf(cnt, 1.0f);

  const float shv  = sheet[i];
  const float open = (shv < 0.1f) ? snode * (0.1f - shv) * 0.5f : 0.0f; // /L_C
  const float eff  = RHOW_G * bed[i] + RHOI_G * ice[i] - p;
  const float sc   = ACLf * shv * relu3(eff);

  __builtin_nontemporal_store(io,   out + OFF_INOUT + i);
  __builtin_nontemporal_store(open, out + OFF_OPEN  + i);
  __builtin_nontemporal_store(sc,   out + OFF_SC    + i);
}

extern "C" void kernel_launch(void* const* d_in, const int* in_sizes, int n_in,
                              void* d_out, int out_size, void* d_ws, size_t ws_size,
                              hipStream_t stream) {
  (void)in_sizes; (void)n_in; (void)out_size; (void)d_ws; (void)ws_size;
  const float* pot   = (const float*)d_in[0];
  const float* sheet = (const float*)d_in[1];
  const float* cs    = (const float*)d_in[2];
  const float* bed   = (const float*)d_in[3];
  const float* ice   = (const float*)d_in[4];
  const float* sv    = (const float*)d_in[5];
  // d_in[6] head, d_in[7] tail, d_in[8] adj are implied by the regular grid.
  const int*   status = (const int*)d_in[9];
  float* out = (float*)d_out;

  dim3 blk(256, 1, 1);
  dim3 grid_rows(8, NYc, 1);                 // 8 * 256 = 2048 covers 2047 links/row
  hlink_kernel<<<grid_rows, blk, 0, stream>>>(pot, sheet, cs, bed, ice, out);

  dim3 grid_v(LVc / (4 * 256), 1, 1);        // 4094 blocks, exact
  vlink_kernel<<<grid_v, blk, 0, stream>>>(pot, sheet, cs, bed, ice, out);

  node_kernel<<<grid_rows, blk, 0, stream>>>(pot, sheet, bed, ice, sv, status, out);
}